// PRNN_1769526526373
// MI455X (gfx1250) — compile-verified
//
#include <hip/hip_runtime.h>
#include <math.h>

#ifndef __has_builtin
#define __has_builtin(x) 0
#endif

#if __has_builtin(__builtin_amdgcn_global_load_async_to_lds_b128)
#define HAVE_ASYNC_LDS 1
#else
#define HAVE_ASYNC_LDS 0
#endif

namespace {
constexpr int kBatch = 8192;
constexpr int kSeq = 128;
constexpr int kNF = 3;
constexpr int kNewton = 12;

constexpr int kRows = 16;          // batch rows per block (= WMMA N dimension)
constexpr int kThreads = 256;      // 8 wave32
constexpr int kWaves = 8;
constexpr int kMatsPerThread = 4;  // 64 matpts/row over 16 threads/row
constexpr int kK = 192;            // NMAT * NF
constexpr int kSigStride = 196;    // padded LDS row stride (floats)
constexpr int kChunksPerWave = 6;  // 48 K-chunks (of 4) over 8 waves
}  // namespace

typedef __attribute__((ext_vector_type(2))) float v2f;
typedef __attribute__((ext_vector_type(8))) float v8f;
typedef int v4i __attribute__((vector_size(16)));

__global__ __launch_bounds__(kThreads)
void prnn_forward_kernel(const float* __restrict__ x,
                         const float* __restrict__ W1,
                         const float* __restrict__ W2,
                         const float* __restrict__ pE,
                         const float* __restrict__ pNu,
                         const float* __restrict__ pA,
                         const float* __restrict__ pB,
                         const float* __restrict__ pC,
                         float* __restrict__ out) {
  __shared__ __align__(16) float xtile[kRows * kSeq * kNF];  // 24.0 KB
  __shared__ __align__(16) float sigT[kRows * kSigStride];   // 12.25 KB
  __shared__ float acc[kWaves * 48];                         // per-wave partial out

  const int tid = threadIdx.x;
  const int lane = tid & 31;
  const int wv = tid >> 5;
  const int b0 = blockIdx.x * kRows;

  // ---- Stage this block's x slice (contiguous 24 KB) into LDS ----
  {
    const float* gsrc = x + (size_t)b0 * kSeq * kNF;
#if HAVE_ASYNC_LDS
    typedef __attribute__((address_space(1))) v4i* gvec_t;
    typedef __attribute__((address_space(3))) v4i* lvec_t;
    gvec_t g = (gvec_t)gsrc;
    lvec_t l = (lvec_t)&xtile[0];
#pragma unroll
    for (int it = 0; it < (kRows * kSeq * kNF) / (4 * kThreads); ++it) {
      const int idx = tid + it * kThreads;  // in 16-byte elements
      __builtin_amdgcn_global_load_async_to_lds_b128(g + idx, l + idx, 0, 0);
    }
    asm volatile("s_wait_asynccnt 0" ::: "memory");
#else
    const float4* s4 = (const float4*)gsrc;
    float4* d4 = (float4*)&xtile[0];
#pragma unroll
    for (int idx = tid; idx < (kRows * kSeq * kNF) / 4; idx += kThreads)
      d4[idx] = s4[idx];
#endif
  }

  // ---- Material constants ----
  const float E = pE[0], nu = pNu[0], A = pA[0], Bp = pB[0], Cp = pC[0];
  const float G = E / (2.0f * (1.0f + nu));
  const float Kb = E / (3.0f * (1.0f - 2.0f * nu));
  const float threeG = 3.0f * G;
  const float twoG = 2.0f * G;
  const float invCp = 1.0f / Cp;
  const float BpOverCp = Bp * invCp;

  // ---- Per-thread material-point mapping ----
  const int r = tid >> 4;  // batch row within tile: 0..15
  const int q = tid & 15;  // sub-slot: matpts p = 4q..4q+3

  // Cache the W1 rows used by this thread's 4 material points (36 floats)
  float w1c[kMatsPerThread][3][3];
#pragma unroll
  for (int i = 0; i < kMatsPerThread; ++i) {
    const int p = 4 * q + i;
#pragma unroll
    for (int c = 0; c < 3; ++c) {
#pragma unroll
      for (int k = 0; k < 3; ++k) w1c[i][c][k] = W1[(3 * p + c) * 3 + k];
    }
  }

  // ---- Time-invariant WMMA A fragments (W2 chunks): out^T = W2 * sig^T ----
  // A is 16x4 fp32: lanes 0-15 hold K=0 (v0) / K=1 (v1); lanes 16-31 K=2 / K=3.
  const int jrow = lane & 15;          // M row = output component j (0..2 used)
  const int khalf = (lane >> 4) * 2;   // 0 or 2
  v2f afrag[kChunksPerWave];
#pragma unroll
  for (int ic = 0; ic < kChunksPerWave; ++ic) {
    const int k0 = wv * (kChunksPerWave * 4) + ic * 4;
    const bool valid = jrow < kNF;
    afrag[ic][0] = valid ? W2[jrow * kK + k0 + khalf] : 0.0f;
    afrag[ic][1] = valid ? W2[jrow * kK + k0 + khalf + 1] : 0.0f;
  }

  // ---- Recurrent state (registers) ----
  float epsp[kMatsPerThread][3];
  float epspeq[kMatsPerThread];
#pragma unroll
  for (int i = 0; i < kMatsPerThread; ++i) {
    epsp[i][0] = 0.0f; epsp[i][1] = 0.0f; epsp[i][2] = 0.0f;
    epspeq[i] = 0.0f;
  }

  __syncthreads();

  for (int t = 0; t < kSeq; ++t) {
    const float xt0 = xtile[r * (kSeq * kNF) + t * kNF + 0];
    const float xt1 = xtile[r * (kSeq * kNF) + t * kNF + 1];
    const float xt2 = xtile[r * (kSeq * kNF) + t * kNF + 2];

    float sg[kMatsPerThread][3];
#pragma unroll
    for (int i = 0; i < kMatsPerThread; ++i) {
      // fc1 for this material point
      const float e0 = w1c[i][0][0] * xt0 + w1c[i][0][1] * xt1 + w1c[i][0][2] * xt2;
      const float e1 = w1c[i][1][0] * xt0 + w1c[i][1][1] * xt1 + w1c[i][1][2] * xt2;
      const float e2 = w1c[i][2][0] * xt0 + w1c[i][2][1] * xt1 + w1c[i][2][2] * xt2;

      const float pxx = epsp[i][0], pyy = epsp[i][1], pxy = epsp[i][2];
      const float peq = epspeq[i];
      const float exx = e0 - pxx;
      const float eyy = e1 - pyy;
      const float gxy = e2 - pxy;
      const float ezz = pxx + pyy;
      const float ev = exx + eyy + ezz;
      const float m = ev * (1.0f / 3.0f);
      const float sxx = twoG * (exx - m);
      const float syy = twoG * (eyy - m);
      const float szz = twoG * (ezz - m);
      const float sxy = G * gxy;
      const float seq =
          sqrtf(1.5f * (sxx * sxx + syy * syy + szz * szz + 2.0f * sxy * sxy));
      const float f = seq - (A + Bp * __expf(-peq * invCp));

      float dgam = 0.0f;
#pragma unroll
      for (int n = 0; n < kNewton; ++n) {
        const float ecur = peq + dgam;
        const float ex = __expf(-ecur * invCp);
        const float phi = seq - threeG * dgam - (A + Bp * ex);
        const float den = threeG - BpOverCp * ex;  // 3G + dsigY(e)
        dgam = fmaxf(dgam + phi / den, 0.0f);
      }
      dgam = (f > 0.0f) ? dgam : 0.0f;
      const float seq_safe = (seq > 0.0f) ? seq : 1.0f;
      const float fac = 1.5f * dgam / seq_safe;
      const float scale = 1.0f - threeG * dgam / seq_safe;
      const float pr = Kb * ev;

      sg[i][0] = sxx * scale + pr;
      sg[i][1] = syy * scale + pr;
      sg[i][2] = sxy * scale;
      epsp[i][0] = pxx + fac * sxx;
      epsp[i][1] = pyy + fac * syy;
      epsp[i][2] = pxy + 2.0f * fac * sxy;
      epspeq[i] = peq + dgam;
    }

    // Write 12 contiguous sig values (cols 12q..12q+11 of this row) as 3x b128
    {
      float4* dst = (float4*)&sigT[r * kSigStride + 12 * q];
      dst[0] = make_float4(sg[0][0], sg[0][1], sg[0][2], sg[1][0]);
      dst[1] = make_float4(sg[1][1], sg[1][2], sg[2][0], sg[2][1]);
      dst[2] = make_float4(sg[2][2], sg[3][0], sg[3][1], sg[3][2]);
    }
    __syncthreads();

    // ---- fc2 partial via V_WMMA_F32_16X16X4_F32 over this wave's K-range ----
    // B is 4x16 fp32: lane L holds B[khalf][N=L%16] (v0), B[khalf+1][N] (v1);
    // B[k][n] = sig[row n][k]  -> ds_load_b64 from the padded sig tile.
    v8f cf = {0.f, 0.f, 0.f, 0.f, 0.f, 0.f, 0.f, 0.f};
#pragma unroll
    for (int ic = 0; ic < kChunksPerWave; ++ic) {
      const int k0 = wv * (kChunksPerWave * 4) + ic * 4;
      const float* bsrc = &sigT[jrow * kSigStride + k0 + khalf];
      v2f bf;
      bf[0] = bsrc[0];
      bf[1] = bsrc[1];
      cf = __builtin_amdgcn_wmma_f32_16x16x4_f32(false, afrag[ic], false, bf,
                                                 (short)0, cf, false, false);
    }
    // D rows 0..2 (lanes 0-15, VGPRs 0..2) are the 3 output components.
    if (lane < 16) {
      acc[wv * 48 + 0 * 16 + lane] = cf[0];
      acc[wv * 48 + 1 * 16 + lane] = cf[1];
      acc[wv * 48 + 2 * 16 + lane] = cf[2];
    }
    __syncthreads();

    // Deterministic cross-wave K-reduction + store out[b][t][j]
    if (tid < 48) {
      float s = 0.0f;
#pragma unroll
      for (int w = 0; w < kWaves; ++w) s += acc[w * 48 + tid];
      const int n = tid & 15;
      const int j = tid >> 4;
      out[((b0 + n) * kSeq + t) * kNF + j] = s;
    }
  }
}

extern "C" void kernel_launch(void* const* d_in, const int* in_sizes, int n_in,
                              void* d_out, int out_size, void* d_ws,
                              size_t ws_size, hipStream_t stream) {
  (void)in_sizes; (void)n_in; (void)out_size; (void)d_ws; (void)ws_size;
  const float* x = (const float*)d_in[0];
  const float* W1 = (const float*)d_in[1];
  const float* W2 = (const float*)d_in[2];
  const float* E = (const float*)d_in[3];
  const float* nu = (const float*)d_in[4];
  const float* A = (const float*)d_in[5];
  const float* Bp = (const float*)d_in[6];
  const float* Cp = (const float*)d_in[7];
  float* out = (float*)d_out;

  dim3 grid(kBatch / kRows);   // 512 blocks
  dim3 block(kThreads);        // 8 wave32 each
  hipLaunchKernelGGL(prnn_forward_kernel, grid, block, 0, stream, x, W1, W2, E,
                     nu, A, Bp, Cp, out);
}